// PatchEmbedding_15994458211143
// MI455X (gfx1250) — compile-verified
//
#include <hip/hip_runtime.h>
#include <hip/hip_bf16.h>

typedef __attribute__((ext_vector_type(2))) float v2f;
typedef __attribute__((ext_vector_type(4))) float v4f;
typedef __attribute__((ext_vector_type(8))) float v8f;

#define N_ROWS  65536
#define MAX_LEN 768
#define D_MODEL 512
#define T_OUT   766   // conv output length (768 - 2)

// ---------------------------------------------------------------------------
// Kernel 1: per-row masked reduction -> tap means, stored as {m0,m1,m2,1.0}.
// The trailing 1.0 is the K=3 pad of the WMMA A-matrix; it multiplies the
// bias row placed in B's K=3 slot, so the GEMM kernel needs no separate
// bias add and can use an inline-0 C operand.
// One wave (32 lanes) per row; 6 x float4 nontemporal loads per lane.
// ---------------------------------------------------------------------------
__global__ __launch_bounds__(256) void tap_means_kernel(const float* __restrict__ x,
                                                        const int* __restrict__ lengths,
                                                        float* __restrict__ m) {
  const int wid  = threadIdx.x >> 5;
  const int lane = threadIdx.x & 31;
  const int row  = blockIdx.x * 8 + wid;

  const v4f* xr = reinterpret_cast<const v4f*>(x + (size_t)row * MAX_LEN);
  const int len = lengths[row];

  float S = 0.f;
  float e0 = 0.f, e1 = 0.f, e766 = 0.f, e767 = 0.f;
#pragma unroll
  for (int i = 0; i < 6; ++i) {
    const int q = lane + 32 * i;          // float4 index within row (0..191)
    const v4f v = __builtin_nontemporal_load(&xr[q]);   // streamed once
    const int base = 4 * q;
    const float a = (base + 0 < len) ? v.x : 0.f;
    const float b = (base + 1 < len) ? v.y : 0.f;
    const float c = (base + 2 < len) ? v.z : 0.f;
    const float d = (base + 3 < len) ? v.w : 0.f;
    S += (a + b) + (c + d);
    if (i == 0 && lane == 0)  { e0 = v.x; e1 = v.y; }   // len >= 3 => never masked
    if (i == 5 && lane == 31) { e766 = c; e767 = d; }   // masked boundary values
  }
  // wave32 butterfly reduction
#pragma unroll
  for (int off = 16; off >= 1; off >>= 1) S += __shfl_xor(S, off, 32);

  const float x0   = __shfl(e0,   0, 32);
  const float x1   = __shfl(e1,   0, 32);
  const float x766 = __shfl(e766, 31, 32);
  const float x767 = __shfl(e767, 31, 32);

  if (lane == 0) {
    const float inv = 1.0f / (float)T_OUT;
    float4 o;
    o.x = (S - x766 - x767) * inv;  // tap 0: t = 0..765
    o.y = (S - x0   - x767) * inv;  // tap 1: t = 1..766
    o.z = (S - x0   - x1)   * inv;  // tap 2: t = 2..767
    o.w = 1.0f;                     // K=3 pad: picks up bias row of B
    reinterpret_cast<float4*>(m)[row] = o;
  }
}

// ---------------------------------------------------------------------------
// Kernel 2: out[N,512] = [m|1] @ [W^T ; bias] via V_WMMA_F32_16X16X4_F32, C=0.
// One wave per 16-row stripe; loops over 32 column tiles reusing A registers.
// B 4x16 layout: v0 = K=0 (lanes 0-15) / K=2 (lanes 16-31);
//                v1 = K=1 (lanes 0-15) / K=3 = bias (lanes 16-31).
// All lane selects are value-cndmasks (no divergent branches; EXEC stays ~0).
// ---------------------------------------------------------------------------
__global__ __launch_bounds__(256) void wmma_gemm_kernel(const float* __restrict__ m,
                                                        const float* __restrict__ w,
                                                        const float* __restrict__ bias,
                                                        float* __restrict__ out) {
  const int wid  = threadIdx.x >> 5;
  const int lane = threadIdx.x & 31;
  const int rt   = blockIdx.x * 8 + wid;   // row tile index (16 rows)
  const int r    = lane & 15;
  const int hi   = lane >> 4;              // 0: K={0,1}, 1: K={2,3}

  // A 16x4 f32: lane L(<16) = row L, K={0,1}; lane L(>=16) = row L-16, K={2,3}.
  // m row stride is 4 floats -> two float2 halves per row ({m0,m1} / {m2,1.0}).
  const float2 ap = reinterpret_cast<const float2*>(m)[(size_t)(rt * 16 + r) * 2 + hi];
  v2f a; a.x = ap.x; a.y = ap.y;

#pragma unroll 1
  for (int ct = 0; ct < D_MODEL / 16; ++ct) {
    const int col = ct * 16 + r;
    // Branch-free B construction (address math + value selects only):
    const float wk02 = w[col * 3 + 2 * hi];   // K=0 (lo lanes) / K=2 (hi lanes)
    const float wk1  = w[col * 3 + 1];        // always in-bounds
    const float bs   = bias[col];
    v2f b;
    b.x = wk02;
    b.y = hi ? bs : wk1;                      // K=3 row carries the bias

    v8f c = {};                               // inline 0 C operand
    // D = A x B + 0   (neg_a, A, neg_b, B, c_mod, C, reuse_a, reuse_b)
    c = __builtin_amdgcn_wmma_f32_16x16x4_f32(false, a, false, b, (short)0, c,
                                              false, false);

    // C/D layout: VGPR j -> row rt*16 + j + 8*hi, col = ct*16 + (lane&15)
    float* o = out + (size_t)(rt * 16 + hi * 8) * D_MODEL + ct * 16 + r;
#pragma unroll
    for (int j = 0; j < 8; ++j)
      __builtin_nontemporal_store(c[j], o + (size_t)j * D_MODEL);
  }
}

extern "C" void kernel_launch(void* const* d_in, const int* in_sizes, int n_in,
                              void* d_out, int out_size, void* d_ws, size_t ws_size,
                              hipStream_t stream) {
  const float* x       = (const float*)d_in[0];   // [65536, 768]
  const int*   lengths = (const int*)  d_in[1];   // [65536]
  const float* conv_w  = (const float*)d_in[2];   // [512, 3]
  const float* conv_b  = (const float*)d_in[3];   // [512]
  float*       out     = (float*)d_out;           // [65536, 512]
  float*       m       = (float*)d_ws;            // [65536, 4] = 1 MiB scratch

  // Phase 1: 8 rows per 256-thread block (wave per row)
  tap_means_kernel<<<N_ROWS / 8, 256, 0, stream>>>(x, lengths, m);
  // Phase 2: 8 row-tiles (of 16 rows) per block, each wave sweeps 32 col tiles
  wmma_gemm_kernel<<<(N_ROWS / 16) / 8, 256, 0, stream>>>(m, conv_w, conv_b, out);
}